// B3SplineUWTApprox2_34797825032658
// MI455X (gfx1250) — compile-verified
//
#include <hip/hip_runtime.h>

typedef float v2f __attribute__((ext_vector_type(2)));
typedef float v4f __attribute__((ext_vector_type(4)));
typedef float v8f __attribute__((ext_vector_type(8)));

constexpr int BATCH   = 16;
constexpr int IMG_H   = 1024;
constexpr int IMG_W   = 1024;
constexpr int LEVELS  = 3;

constexpr int TILE    = 64;               // output tile per workgroup
constexpr int REG     = 96;               // computed region = TILE + 2*16 halo
constexpr int PAD     = 8;                // physical pad ring for WMMA 32-wide A/B windows
constexpr int ROWS    = REG + 2 * PAD;    // 112 physical rows
constexpr int PITCH   = 116;              // row stride in floats (mult of 4, de-banked vs 112)
constexpr int BUFSZ   = ROWS * PITCH;     // floats per buffer (12992)
constexpr int NTHREADS = 384;             // 12 wave32
constexpr int NWAVES  = NTHREADS / 32;
constexpr int TPR     = REG / 16;         // 6 tiles per region row/col

__device__ __forceinline__ int reflect_i(int g, int n) {
  if (g < 0) g = -g;
  if (g >= n) g = 2 * n - 2 - g;
  return g;
}

// weight of the dilated B3 band matrix at offset rel; nonzero iff rel == (k-2)*d
__device__ __forceinline__ float band_wt(int rel, int d, int lvl) {
  float w = 0.0f;
  if ((rel & (d - 1)) == 0) {
    int a = rel < 0 ? -rel : rel;
    int q = a >> lvl;               // |rel| / d
    if (q == 0)      w = 0.375f;
    else if (q == 1) w = 0.25f;
    else if (q == 2) w = 0.0625f;
  }
  return w;
}

__global__ __launch_bounds__(NTHREADS)
void uwt_b3_kernel(const float* __restrict__ x, float* __restrict__ out) {
  extern __shared__ float smem[];

  const int tid  = threadIdx.x;
  const int lane = tid & 31;
  const int wave = tid >> 5;               // 0..11
  const int gw0  = blockIdx.x * TILE;
  const int gh0  = blockIdx.y * TILE;
  const int b    = blockIdx.z;

  // wave -> tile-column, tile-row block (computed once; wave-uniform)
  const int tc  = wave % TPR;              // 0..5
  const int trb = (wave / TPR) * 3;        // 0 or 3; tiles tr = trb + i, i = 0..2

  // lane-invariant WMMA fragment coordinates (wave32 f32 16x16x4 layout)
  const int nn = lane & 15;               // A: row M / B,D: col N
  const int kb = (lane & 16) ? 2 : 0;     // K sub-block for A/B fragments
  const int rb = (lane & 16) ? 8 : 0;     // D row base

  // ---- zero only the padding rings of the three buffers ----
  {
    v4f z = {0.f, 0.f, 0.f, 0.f};
#pragma unroll
    for (int bu = 0; bu < 3; ++bu) {
      v4f* base = reinterpret_cast<v4f*>(smem + bu * BUFSZ);
      // rows 0..7 and 104..111, full pitch
      for (int i = tid; i < (PAD * PITCH) / 4; i += NTHREADS) {
        base[i] = z;
        base[(104 * PITCH) / 4 + i] = z;
      }
      // rows 8..103: side cols 0..7 and 104..111
      for (int i = tid; i < 96 * 4; i += NTHREADS) {
        int r = 8 + (i >> 2);
        int sel = i & 3;
        int c = (sel < 2) ? (sel << 2) : (104 + ((sel - 2) << 2));
        *reinterpret_cast<v4f*>(smem + bu * BUFSZ + r * PITCH + c) = z;
      }
    }
  }
  __syncthreads();

  // ---- load 96x96 reflect-padded region into buffer A ----
  {
    const float* xb = x + (size_t)b * IMG_H * IMG_W;
    float* bufA = smem;
    for (int i = tid; i < REG * REG; i += NTHREADS) {   // 24 uniform iterations
      int r  = i / REG, c = i % REG;
      int gh = reflect_i(gh0 - 16 + r, IMG_H);
      int gw = reflect_i(gw0 - 16 + c, IMG_W);
      bufA[(r + PAD) * PITCH + (c + PAD)] = xb[(size_t)gh * IMG_W + gw];
    }
  }
  __syncthreads();

  float* X = smem;                // current approximation
  float* T = smem + BUFSZ;        // temp after H-pass
  float* S = smem + 2 * BUFSZ;    // smoothed result

  // level-3 tile ranges: s3 only consumed on central 64x64
  constexpr int h_lo[LEVELS] = {0, 0, 1};   // H-pass tr range [lo, hi)
  constexpr int h_hi[LEVELS] = {6, 6, 5};
  constexpr int w_lo[LEVELS] = {0, 0, 1};   // W-pass tr AND tc range [lo, hi)
  constexpr int w_hi[LEVELS] = {6, 6, 5};

#pragma unroll
  for (int lvl = 0; lvl < LEVELS; ++lvl) {
    const int d = 1 << lvl;

    // banded dilated-tap fragments: identical values serve the H-pass (as A)
    // and the W-pass (as B) by symmetry of the band formula.
    v2f wf[8];
#pragma unroll
    for (int t = 0; t < 8; ++t) {
      wf[t].x = band_wt(4 * t + kb + 0 - 8 - nn, d, lvl);
      wf[t].y = band_wt(4 * t + kb + 1 - 8 - nn, d, lvl);
    }

    // ---- pass 1: conv along H.  T_tile(16x16) = Mh(16x32) x X_window(32x16) ----
    {
      const int col = tc * 16 + PAD + nn;
#pragma unroll
      for (int i = 0; i < 3; ++i) {
        const int tr = trb + i;                       // wave-uniform per unroll iter
        if (tr < h_lo[lvl] || tr >= h_hi[lvl]) continue;   // wave-uniform skip
        v8f acc = {0.f, 0.f, 0.f, 0.f, 0.f, 0.f, 0.f, 0.f};
#pragma unroll
        for (int t = 0; t < 8; ++t) {
          const int row = tr * 16 + 4 * t + kb;       // phys row (region row - 8 + PAD)
          v2f bfrag;
          bfrag.x = X[row * PITCH + col];
          bfrag.y = X[(row + 1) * PITCH + col];
          acc = __builtin_amdgcn_wmma_f32_16x16x4_f32(
              false, wf[t], false, bfrag, (short)0, acc, false, false);
        }
        const int orow = tr * 16 + PAD + rb;
#pragma unroll
        for (int v = 0; v < 8; ++v) T[(orow + v) * PITCH + col] = acc[v];
      }
    }
    __syncthreads();

    // ---- pass 2: conv along W.  S_tile(16x16) = T_window(16x32) x Mw(32x16) ----
    if (tc >= w_lo[lvl] && tc < w_hi[lvl]) {          // wave-uniform: EXEC stays full
      const int ocol = tc * 16 + PAD + nn;
#pragma unroll
      for (int i = 0; i < 3; ++i) {
        const int tr = trb + i;
        if (tr < w_lo[lvl] || tr >= w_hi[lvl]) continue;
        const int row = tr * 16 + PAD + nn;           // A fragment: lane row = M
        v8f acc = {0.f, 0.f, 0.f, 0.f, 0.f, 0.f, 0.f, 0.f};
#pragma unroll
        for (int t = 0; t < 8; ++t) {
          const int colk = tc * 16 + 4 * t + kb;      // phys col (region col - 8 + PAD)
          v2f afrag = *reinterpret_cast<const v2f*>(&T[row * PITCH + colk]);
          acc = __builtin_amdgcn_wmma_f32_16x16x4_f32(
              false, afrag, false, wf[t], (short)0, acc, false, false);
        }
        const int orow = tr * 16 + PAD + rb;
#pragma unroll
        for (int v = 0; v < 8; ++v) S[(orow + v) * PITCH + ocol] = acc[v];
      }
    }
    __syncthreads();

    // ---- outputs: w_{lvl+1} = X - S over central 64x64 (phys offset 24) ----
    {
      float* outw = out + (((size_t)b * (LEVELS + 1) + lvl) * IMG_H + gh0) * IMG_W + gw0;
      float* outc = out + (((size_t)b * (LEVELS + 1) + LEVELS) * IMG_H + gh0) * IMG_W + gw0;
      for (int i = tid; i < (TILE * TILE) / 4; i += NTHREADS) {
        const int r = i >> 4;
        const int c = (i & 15) << 2;
        const int off = (24 + r) * PITCH + 24 + c;
        v4f xv = *reinterpret_cast<const v4f*>(&X[off]);
        v4f sv = *reinterpret_cast<const v4f*>(&S[off]);
        v4f wv = xv - sv;
        *reinterpret_cast<v4f*>(&outw[(size_t)r * IMG_W + c]) = wv;
        if (lvl == LEVELS - 1)
          *reinterpret_cast<v4f*>(&outc[(size_t)r * IMG_W + c]) = sv;   // c_J
      }
    }

    // rotate: new approximation is S; old X becomes scratch for next S
    float* oldX = X;
    X = S;
    S = oldX;
    __syncthreads();
  }
}

extern "C" void kernel_launch(void* const* d_in, const int* in_sizes, int n_in,
                              void* d_out, int out_size, void* d_ws, size_t ws_size,
                              hipStream_t stream) {
  (void)in_sizes; (void)n_in; (void)d_ws; (void)ws_size; (void)out_size;
  const float* x = reinterpret_cast<const float*>(d_in[0]);
  float* out = reinterpret_cast<float*>(d_out);

  dim3 grid(IMG_W / TILE, IMG_H / TILE, BATCH);   // 16 x 16 x 16
  dim3 block(NTHREADS);
  size_t shmem = (size_t)3 * BUFSZ * sizeof(float);   // 155,904 B (2 blocks/WGP at 320 KB)
  uwt_b3_kernel<<<grid, block, shmem, stream>>>(x, out);
}